// FeatureDecoder_87729001988299
// MI455X (gfx1250) — compile-verified
//
#include <hip/hip_runtime.h>

typedef __attribute__((ext_vector_type(2))) float v2f;
typedef __attribute__((ext_vector_type(8))) float v8f;

#define LATENT 128
#define NUM_CLASSES 10
#define TEACHER 96
#define FEAT 256
#define HIDDEN 128
#define ZK 144        // concat dim 138 padded to multiple of 4 (and 16B)
#define ZK_REAL 138

__device__ __forceinline__ v8f wmma_f32_step(v2f a, v2f b, v8f c) {
  // D(16x16,f32) = A(16x4,f32) * B(4x16,f32) + C
  return __builtin_amdgcn_wmma_f32_16x16x4_f32(
      /*neg_a=*/false, a, /*neg_b=*/false, b,
      /*c_mod=*/(short)0, c, /*reuse_a=*/false, /*reuse_b=*/false);
}

// ---------------------------------------------------------------------------
// Kernel 1: fused MLP.  h = relu([z|onehot] @ W1^T + b1); z_feat = h @ W2^T + b2
// One block = 16 rows, 256 threads (8 waves). Wave w owns output cols [16w,16w+16).
// ---------------------------------------------------------------------------
__global__ void __launch_bounds__(256) mlp_kernel(
    const float* __restrict__ z_nodes, const int* __restrict__ y_pred,
    const float* __restrict__ W1, const float* __restrict__ b1,
    const float* __restrict__ W2, const float* __restrict__ b2,
    float* __restrict__ z_feat) {
  __shared__ float zt[16 * ZK];       // padded input tile (zeros beyond 138)
  __shared__ float ht[16 * HIDDEN];   // hidden tile
  const int row0 = blockIdx.x * 16;
  const int tid  = threadIdx.x;

  for (int i = tid; i < 16 * ZK; i += 256) {
    const int r = i / ZK, k = i % ZK;
    float v;
    if (k < LATENT)        v = z_nodes[(row0 + r) * LATENT + k];
    else if (k < ZK_REAL)  v = (y_pred[row0 + r] == (k - LATENT)) ? 1.0f : 0.0f;
    else                   v = 0.0f;
    zt[i] = v;
  }
  __syncthreads();

  const int lane = tid & 31;
  const int wave = tid >> 5;
  const int half = lane >> 4;   // which 16-lane half supplies K pair 0/1 vs 2/3
  const int mr   = lane & 15;

  // MM1: 128 output cols = 8 waves x 16 cols, K = 144 (padded)
  {
    const int n0 = wave * 16;
    v8f c = {};
    for (int k0 = 0; k0 < ZK; k0 += 4) {
      const int k = k0 + 2 * half;
      v2f a, b;
      a.x = zt[mr * ZK + k];
      a.y = zt[mr * ZK + k + 1];
      if (k < ZK_REAL) {                    // A is zero-padded, guard only W1 reads
        b.x = W1[(n0 + mr) * ZK_REAL + k];
        b.y = W1[(n0 + mr) * ZK_REAL + k + 1];
      } else { b.x = 0.0f; b.y = 0.0f; }
      c = wmma_f32_step(a, b, c);
    }
    const float bias = b1[n0 + mr];
    for (int r = 0; r < 8; ++r) {
      const float v = c[r] + bias;
      ht[(r + 8 * half) * HIDDEN + n0 + mr] = v > 0.0f ? v : 0.0f;  // relu
    }
  }
  __syncthreads();

  // MM2: 96 output cols = waves 0..5, K = 128
  if (wave < TEACHER / 16) {
    const int n0 = wave * 16;
    v8f c = {};
    for (int k0 = 0; k0 < HIDDEN; k0 += 4) {
      const int k = k0 + 2 * half;
      v2f a, b;
      a.x = ht[mr * HIDDEN + k];
      a.y = ht[mr * HIDDEN + k + 1];
      b.x = W2[(n0 + mr) * HIDDEN + k];
      b.y = W2[(n0 + mr) * HIDDEN + k + 1];
      c = wmma_f32_step(a, b, c);
    }
    const float bias = b2[n0 + mr];
    for (int r = 0; r < 8; ++r)
      z_feat[(row0 + r + 8 * half) * TEACHER + n0 + mr] = c[r] + bias;
  }
}

// ---------------------------------------------------------------------------
// Generic WMMA GEMM: Y[16 rows x NCOLS] = X @ W^T (+ bias).
// blockDim = 32 * (NCOLS/16); one 16-col tile per wave; A tile staged in LDS.
// ---------------------------------------------------------------------------
template <int KDIM, int NCOLS, bool BIAS>
__global__ void gemm_wmma_kernel(const float* __restrict__ X,
                                 const float* __restrict__ W,
                                 const float* __restrict__ bias,
                                 float* __restrict__ Y) {
  __shared__ float xt[16 * KDIM];
  const int row0 = blockIdx.x * 16;
  const int tid  = threadIdx.x;
  constexpr int NT = 32 * (NCOLS / 16);
  for (int i = tid; i < 16 * KDIM; i += NT) xt[i] = X[row0 * KDIM + i];
  __syncthreads();

  const int lane = tid & 31, wave = tid >> 5;
  const int half = lane >> 4, mr = lane & 15;
  const int n0 = wave * 16;
  v8f c = {};
  for (int k0 = 0; k0 < KDIM; k0 += 4) {
    const int k = k0 + 2 * half;
    v2f a, b;
    a.x = xt[mr * KDIM + k];
    a.y = xt[mr * KDIM + k + 1];
    b.x = W[(n0 + mr) * KDIM + k];
    b.y = W[(n0 + mr) * KDIM + k + 1];
    c = wmma_f32_step(a, b, c);
  }
  float bv = 0.0f;
  if constexpr (BIAS) bv = bias[n0 + mr];
  for (int r = 0; r < 8; ++r)
    Y[(row0 + r + 8 * half) * NCOLS + n0 + mr] = c[r] + bv;
}

// ---------------------------------------------------------------------------
// GCN aggregation helpers
// ---------------------------------------------------------------------------
__global__ void deg_init_kernel(float* __restrict__ deg, int n) {
  const int t = blockIdx.x * blockDim.x + threadIdx.x;
  if (t < n) deg[t] = 1.0f;  // self-loop
}

__global__ void deg_edge_kernel(const int* __restrict__ ei, float* __restrict__ deg, int E) {
  const int t = blockIdx.x * blockDim.x + threadIdx.x;
  if (t < E) atomicAdd(&deg[ei[E + t]], 1.0f);  // dst row of edge_index
}

// dinv = rsqrt(deg); accum initialized with the self-loop term xw * dinv^2
__global__ void dinv_accum_kernel(const float* __restrict__ deg,
                                  const float* __restrict__ xw,
                                  float* __restrict__ dinv,
                                  float* __restrict__ accum, int n) {
  const int t = blockIdx.x * blockDim.x + threadIdx.x;
  if (t >= n * (TEACHER / 4)) return;
  const int node = t / (TEACHER / 4);
  const int ch   = t % (TEACHER / 4);
  const float di = rsqrtf(deg[node]);   // deg >= 1 always
  if (ch == 0) dinv[node] = di;
  const float s = di * di;
  const float4 v = *(const float4*)(xw + (size_t)node * TEACHER + ch * 4);
  float4 o; o.x = v.x * s; o.y = v.y * s; o.z = v.z * s; o.w = v.w * s;
  *(float4*)(accum + (size_t)node * TEACHER + ch * 4) = o;
}

// accum[dst] += xw[src] * dinv[src]*dinv[dst], 4 features per thread
__global__ void scatter_kernel(const int* __restrict__ ei,
                               const float* __restrict__ dinv,
                               const float* __restrict__ xw,
                               float* __restrict__ accum, int E) {
  const int t = blockIdx.x * blockDim.x + threadIdx.x;
  const int e  = t / (TEACHER / 4);
  const int ch = t % (TEACHER / 4);
  if (e >= E) return;
  const int s = ei[e];
  const int d = ei[E + e];
  const float nrm = dinv[s] * dinv[d];
  const float4 v = *(const float4*)(xw + (size_t)s * TEACHER + ch * 4);
  float* o = accum + (size_t)d * TEACHER + ch * 4;
  atomicAdd(o + 0, v.x * nrm);
  atomicAdd(o + 1, v.y * nrm);
  atomicAdd(o + 2, v.z * nrm);
  atomicAdd(o + 3, v.w * nrm);
}

__global__ void finalize_kernel(const float* __restrict__ accum,
                                const float* __restrict__ gcn_b,
                                float* __restrict__ g, int total) {
  const int t = blockIdx.x * blockDim.x + threadIdx.x;
  if (t < total) {
    const float v = accum[t] + gcn_b[t % TEACHER];
    g[t] = v > 0.0f ? v : 0.0f;  // relu
  }
}

// ---------------------------------------------------------------------------
extern "C" void kernel_launch(void* const* d_in, const int* in_sizes, int n_in,
                              void* d_out, int out_size, void* d_ws, size_t ws_size,
                              hipStream_t stream) {
  const float* z_nodes = (const float*)d_in[0];
  const int*   y_pred  = (const int*)d_in[1];
  const int*   ei      = (const int*)d_in[2];
  const float* W1      = (const float*)d_in[3];
  const float* b1      = (const float*)d_in[4];
  const float* W2      = (const float*)d_in[5];
  const float* b2      = (const float*)d_in[6];
  const float* gcnW    = (const float*)d_in[7];
  const float* gcnb    = (const float*)d_in[8];
  const float* Wf      = (const float*)d_in[9];
  const float* bf      = (const float*)d_in[10];
  float* out = (float*)d_out;

  const int N = in_sizes[0] / LATENT;  // 50000 (multiple of 16)
  const int E = in_sizes[2] / 2;       // 800000

  float* ws     = (float*)d_ws;
  float* z_feat = ws;                            // N*96 (reused as g after GCN)
  float* xw     = ws + (size_t)N * TEACHER;      // N*96
  float* accum  = ws + 2 * (size_t)N * TEACHER;  // N*96
  float* deg    = ws + 3 * (size_t)N * TEACHER;  // N
  float* dinv   = deg + N;                       // N

  const int nblk16 = N / 16;

  // 1. MLP -> z_feat
  mlp_kernel<<<nblk16, 256, 0, stream>>>(z_nodes, y_pred, W1, b1, W2, b2, z_feat);
  // 2. GCN linear: xw = z_feat @ gcn_W^T (bias applied after aggregation)
  gemm_wmma_kernel<TEACHER, TEACHER, false>
      <<<nblk16, 32 * (TEACHER / 16), 0, stream>>>(z_feat, gcnW, nullptr, xw);
  // 3-5. degrees, dinv, self-loop-initialized accumulator
  deg_init_kernel<<<(N + 255) / 256, 256, 0, stream>>>(deg, N);
  deg_edge_kernel<<<(E + 255) / 256, 256, 0, stream>>>(ei, deg, E);
  dinv_accum_kernel<<<(N * (TEACHER / 4) + 255) / 256, 256, 0, stream>>>(deg, xw, dinv, accum, N);
  // 6. edge scatter-add
  scatter_kernel<<<((size_t)E * (TEACHER / 4) + 255) / 256, 256, 0, stream>>>(ei, dinv, xw, accum, E);
  // 7. bias + relu -> g (reuse z_feat buffer)
  finalize_kernel<<<(N * TEACHER + 255) / 256, 256, 0, stream>>>(accum, gcnb, z_feat, N * TEACHER);
  // 8. final projection: out = g @ Wf^T + bf  -> [N, 256]
  gemm_wmma_kernel<TEACHER, FEAT, true>
      <<<nblk16, 32 * (FEAT / 16), 0, stream>>>(z_feat, Wf, bf, out);
}